// StandaloneMELayerWithBinding_54752243090063
// MI455X (gfx1250) — compile-verified
//
#include <hip/hip_runtime.h>
#include <stdint.h>

#define DIMN     2048
#define KTOP     8
#define NPAIR    28
#define MTR      16
#define TSTEPS   256
#define NTHREADS 256
#define EPT      8          // elements (dims) per thread: 2048/256
#define NWAVES   (NTHREADS / 32)
#define LIFETIME 5

// ---------------------------------------------------------------------------
// gfx1250 async global->LDS copy of 32 bytes per lane (two b128 chunks).
// INST_OFFSET applies to BOTH the LDS and global address per CDNA5 ISA
// so one base pair suffices.
// Tracked by ASYNCcnt; wave-private consumption => s_wait_asynccnt only.
// ---------------------------------------------------------------------------
__device__ __forceinline__ void async_load_32B(const float* g, float* lds) {
  uint32_t l  = (uint32_t)(uintptr_t)lds;   // low 32 bits of flat LDS addr == LDS offset
  uint64_t ga = (uint64_t)(uintptr_t)g;
  asm volatile("global_load_async_to_lds_b128 %0, %1, off"           :: "v"(l), "v"(ga) : "memory");
  asm volatile("global_load_async_to_lds_b128 %0, %1, off offset:16" :: "v"(l), "v"(ga) : "memory");
}

__device__ __forceinline__ void wait_async_zero() {
  asm volatile("s_wait_asynccnt 0" ::: "memory");
}

__global__ __launch_bounds__(NTHREADS) void me_binding_kernel(
    const float* __restrict__ x,
    const float* __restrict__ tape_re,
    const float* __restrict__ tape_im,
    const float* __restrict__ eta_p,
    const float* __restrict__ tb_re_g,
    const float* __restrict__ tb_im_g,
    float* __restrict__ out)
{
  __shared__ __align__(16) float hbuf[2][DIMN];   // double-buffered x row
  __shared__ float wredv[NWAVES];
  __shared__ int   wredi[NWAVES];
  __shared__ float bcast;
  __shared__ int   topIdx[KTOP];
  __shared__ float topCre[KTOP], topCim[KTOP], topSre[KTOP], topSim[KTOP];
  __shared__ int   tiS[MTR], tjS[MTR], cntS[MTR];
  __shared__ float mreS[MTR], mimS[MTR];

  const int tid  = threadIdx.x;
  const int lane = tid & 31;
  const int wave = tid >> 5;
  const int b    = blockIdx.x;
  const int base = tid * EPT;

  // ---- init per-block transient state ----
  if (tid < MTR) { tiS[tid] = 0; tjS[tid] = 0; cntS[tid] = 0; mreS[tid] = 0.f; mimS[tid] = 0.f; }

  // ---- load tape + torque bias, compute s0 = renorm(tape) ----
  float s_re[EPT], s_im[EPT], tbr[EPT], tbi[EPT];
  float sums = 0.f;
  #pragma unroll
  for (int j = 0; j < EPT; ++j) {
    float r  = tape_re[base + j];
    float im = tape_im[base + j];
    s_re[j] = r; s_im[j] = im;
    tbr[j] = tb_re_g[base + j];
    tbi[j] = tb_im_g[base + j];
    sums += r * r + im * im;
  }
  #pragma unroll
  for (int off = 16; off > 0; off >>= 1) sums += __shfl_down(sums, off, 32);
  if (lane == 0) wredv[wave] = sums;
  __syncthreads();
  if (tid == 0) { float a = 0.f; for (int w = 0; w < NWAVES; ++w) a += wredv[w]; bcast = a; }
  __syncthreads();
  {
    float inv = 1.0f / fmaxf(sqrtf(bcast), 1e-8f);
    #pragma unroll
    for (int j = 0; j < EPT; ++j) { s_re[j] *= inv; s_im[j] *= inv; }
  }
  const float eta = fabsf(eta_p[0]);

  const float* xb = x   + (size_t)b * TSTEPS * DIMN;
  float*       ob = out + (size_t)b * TSTEPS * DIMN;

  // ---- prologue: async-stream row 0 into hbuf[0] ----
  async_load_32B(xb + base, &hbuf[0][base]);

  for (int t = 0; t < TSTEPS; ++t) {
    const int cur = t & 1;

    // wait for this wave's async copy of row t, read h, kick off row t+1
    wait_async_zero();
    float h[EPT];
    {
      float4 h0 = *(const float4*)&hbuf[cur][base];
      float4 h1 = *(const float4*)&hbuf[cur][base + 4];
      h[0] = h0.x; h[1] = h0.y; h[2] = h0.z; h[3] = h0.w;
      h[4] = h1.x; h[5] = h1.y; h[6] = h1.z; h[7] = h1.w;
    }
    if (t + 1 < TSTEPS)
      async_load_32B(xb + (size_t)(t + 1) * DIMN + base, &hbuf[cur ^ 1][base]);

    // ---- c = h * s; rank by |c|^2 (same ordering as |c|) ----
    float c_re[EPT], c_im[EPT], mag2[EPT];
    #pragma unroll
    for (int j = 0; j < EPT; ++j) {
      c_re[j] = h[j] * s_re[j];
      c_im[j] = h[j] * s_im[j];
      mag2[j] = c_re[j] * c_re[j] + c_im[j] * c_im[j];
    }

    // ---- top-8 by iterative argmax (tie -> lower index, matches lax.top_k) ----
    #pragma unroll 1
    for (int k = 0; k < KTOP; ++k) {
      float bv = -1.f; int bi = 0;
      #pragma unroll
      for (int j = 0; j < EPT; ++j) {
        float v = mag2[j]; int idx = base + j;
        if (v > bv || (v == bv && idx < bi)) { bv = v; bi = idx; }
      }
      #pragma unroll
      for (int off = 16; off > 0; off >>= 1) {
        float ov = __shfl_down(bv, off, 32);
        int   oi = __shfl_down(bi, off, 32);
        if (ov > bv || (ov == bv && oi < bi)) { bv = ov; bi = oi; }
      }
      if (lane == 0) { wredv[wave] = bv; wredi[wave] = bi; }
      __syncthreads();
      if (tid == 0) {
        float gv = wredv[0]; int gi = wredi[0];
        for (int w = 1; w < NWAVES; ++w) {
          float ov = wredv[w]; int oi = wredi[w];
          if (ov > gv || (ov == gv && oi < gi)) { gv = ov; gi = oi; }
        }
        topIdx[k] = gi;
      }
      __syncthreads();
      int gi = topIdx[k];
      #pragma unroll
      for (int j = 0; j < EPT; ++j) {
        if (base + j == gi) {
          mag2[j] = -1.f;                     // mask for next extraction
          topCre[k] = c_re[j]; topCim[k] = c_im[j];
          topSre[k] = s_re[j]; topSim[k] = s_im[j];
        }
      }
      __syncthreads();
    }

    // ---- serial bind logic + transient state machine (thread 0) ----
    if (tid == 0) {
      float tcr[KTOP], tci[KTOP], tsr[KTOP], tsi[KTOP]; int tix[KTOP];
      #pragma unroll
      for (int k = 0; k < KTOP; ++k) {
        tcr[k] = topCre[k]; tci[k] = topCim[k];
        tsr[k] = topSre[k]; tsi[k] = topSim[k];
        tix[k] = topIdx[k];
      }
      float score[NPAIR], mrP[NPAIR], miP[NPAIR];
      int   ciP[NPAIR], cjP[NPAIR];
      int   npos = 0, p = 0;
      #pragma unroll
      for (int i = 0; i < KTOP; ++i) {
        #pragma unroll
        for (int j = i + 1; j < KTOP; ++j) {
          // tm_i*tm_j*cos(ph_i-ph_j) == Re(c_i)*Re(c_j)+Im(c_i)*Im(c_j)
          float sc = tcr[i] * tcr[j] + tci[i] * tci[j];
          score[p] = sc; if (sc > 0.f) npos++;
          ciP[p] = tix[i]; cjP[p] = tix[j];
          float pr  = tsr[i] * tsr[j] - tsi[i] * tsi[j];
          float pi2 = tsr[i] * tsi[j] + tsi[i] * tsr[j];
          float den = fmaxf(sqrtf(pr * pr + pi2 * pi2), 1e-8f);
          mrP[p] = 0.05f * pr / den; miP[p] = 0.05f * pi2 / den;
          ++p;
        }
      }
      int n_bind = (int)((float)npos * 0.15f); if (n_bind < 1) n_bind = 1;
      float arr[NPAIR];
      for (int q = 0; q < NPAIR; ++q) arr[q] = (score[q] > 0.f) ? score[q] : -3.0e38f;
      for (int a = 1; a < NPAIR; ++a) {              // insertion sort, descending
        float key = arr[a]; int bq = a - 1;
        while (bq >= 0 && arr[bq] < key) { arr[bq + 1] = arr[bq]; --bq; }
        arr[bq + 1] = key;
      }
      int thidx = npos - 1; if (thidx < 0) thidx = 0;
      if (n_bind - 1 < thidx) thidx = n_bind - 1;
      float theta = arr[thidx];

      int   ti[MTR], tj[MTR], cnt[MTR]; float mre[MTR], mim[MTR];
      #pragma unroll
      for (int k2 = 0; k2 < MTR; ++k2) {
        ti[k2] = tiS[k2]; tj[k2] = tjS[k2]; cnt[k2] = cntS[k2];
        mre[k2] = mreS[k2]; mim[k2] = mimS[k2];
      }
      for (int q = 0; q < NPAIR; ++q) {
        bool dob = (npos > 0) && (score[q] >= theta);
        if (dob) {
          int midx = -1, fidx = -1;
          #pragma unroll
          for (int k2 = 0; k2 < MTR; ++k2) {
            bool active = cnt[k2] > 0;
            if (midx < 0 && active &&
                ((ti[k2] == ciP[q] && tj[k2] == cjP[q]) ||
                 (ti[k2] == cjP[q] && tj[k2] == ciP[q]))) midx = k2;
            if (fidx < 0 && !active) fidx = k2;
          }
          if (midx >= 0) cnt[midx] = LIFETIME;             // refresh
          else if (fidx >= 0) {                            // insert
            ti[fidx] = ciP[q]; tj[fidx] = cjP[q];
            mre[fidx] = mrP[q]; mim[fidx] = miP[q];
            cnt[fidx] = LIFETIME;
          }
        }
      }
      #pragma unroll
      for (int k2 = 0; k2 < MTR; ++k2) {                   // decay + cull
        mre[k2] *= 0.9f; mim[k2] *= 0.9f; cnt[k2] -= 1;
        bool alive = (cnt[k2] > 0) &&
                     (sqrtf(mre[k2] * mre[k2] + mim[k2] * mim[k2]) > 1e-6f);
        if (!alive) { cnt[k2] = 0; mre[k2] = 0.f; mim[k2] = 0.f; }
        tiS[k2] = ti[k2]; tjS[k2] = tj[k2]; cntS[k2] = cnt[k2];
        mreS[k2] = mre[k2]; mimS[k2] = mim[k2];
      }
    }
    __syncthreads();

    // ---- scatter-add transients into s (both endpoints), update, renorm ----
    int   tiL[MTR], tjL[MTR]; float dreL[MTR], dimL[MTR];
    #pragma unroll
    for (int k2 = 0; k2 < MTR; ++k2) {
      tiL[k2] = tiS[k2]; tjL[k2] = tjS[k2];
      dreL[k2] = 0.1f * mreS[k2]; dimL[k2] = 0.1f * mimS[k2];
    }

    float acc = 0.f;
    float ns_re[EPT], ns_im[EPT];
    #pragma unroll
    for (int j = 0; j < EPT; ++j) {
      int g = base + j;
      float ar = 0.f, ai = 0.f;
      #pragma unroll
      for (int k2 = 0; k2 < MTR; ++k2) {
        if (tiL[k2] == g) { ar += dreL[k2]; ai += dimL[k2]; }
        if (tjL[k2] == g) { ar += dreL[k2]; ai += dimL[k2]; }
      }
      float sar = s_re[j] + ar, sai = s_im[j] + ai;        // s_aug
      float car = h[j] * sar,  cai = h[j] * sai;           // c_aug
      bool res = (car >  1e-6f) && (fabsf(cai) <  car);
      bool tor = (car < -1e-6f) || (fabsf(cai) >= fabsf(car));
      float nm = (res || tor) ? 1.f : 0.f;                 // 1 - orth_mask
      float tm = tor ? 1.f : 0.f;
      float ur = eta * (car * nm + tm * tbr[j]);
      float ui = eta * (cai * nm + tm * tbi[j]);
      ns_re[j] = s_re[j] + ur;                             // base s, not s_aug!
      ns_im[j] = s_im[j] + ui;
      acc += ns_re[j] * ns_re[j] + ns_im[j] * ns_im[j];
    }
    #pragma unroll
    for (int off = 16; off > 0; off >>= 1) acc += __shfl_down(acc, off, 32);
    if (lane == 0) wredv[wave] = acc;
    __syncthreads();
    if (tid == 0) { float a = 0.f; for (int w = 0; w < NWAVES; ++w) a += wredv[w]; bcast = a; }
    __syncthreads();
    {
      float inv = 1.0f / fmaxf(sqrtf(bcast), 1e-8f);
      #pragma unroll
      for (int j = 0; j < EPT; ++j) { s_re[j] = ns_re[j] * inv; s_im[j] = ns_im[j] * inv; }
    }

    // ---- out[b,t,:] = |s_new| ----
    float4 o0, o1;
    o0.x = sqrtf(s_re[0]*s_re[0] + s_im[0]*s_im[0]);
    o0.y = sqrtf(s_re[1]*s_re[1] + s_im[1]*s_im[1]);
    o0.z = sqrtf(s_re[2]*s_re[2] + s_im[2]*s_im[2]);
    o0.w = sqrtf(s_re[3]*s_re[3] + s_im[3]*s_im[3]);
    o1.x = sqrtf(s_re[4]*s_re[4] + s_im[4]*s_im[4]);
    o1.y = sqrtf(s_re[5]*s_re[5] + s_im[5]*s_im[5]);
    o1.z = sqrtf(s_re[6]*s_re[6] + s_im[6]*s_im[6]);
    o1.w = sqrtf(s_re[7]*s_re[7] + s_im[7]*s_im[7]);
    *(float4*)&ob[(size_t)t * DIMN + base]     = o0;
    *(float4*)&ob[(size_t)t * DIMN + base + 4] = o1;
  }
}

extern "C" void kernel_launch(void* const* d_in, const int* in_sizes, int n_in,
                              void* d_out, int out_size, void* d_ws, size_t ws_size,
                              hipStream_t stream) {
  const float* x    = (const float*)d_in[0];
  const float* tre  = (const float*)d_in[1];
  const float* tim  = (const float*)d_in[2];
  const float* etap = (const float*)d_in[3];
  const float* tbr  = (const float*)d_in[4];
  const float* tbi  = (const float*)d_in[5];
  float* out = (float*)d_out;

  const int B = in_sizes[0] / (TSTEPS * DIMN);   // 64 for reference shapes
  me_binding_kernel<<<dim3(B), dim3(NTHREADS), 0, stream>>>(
      x, tre, tim, etap, tbr, tbi, out);
}